// Discounted_Cumsum_46643344835363
// MI455X (gfx1250) — compile-verified
//
#include <hip/hip_runtime.h>

// Discounted cumulative sum y[t] = x[t] + gamma[h]*y[t-1] over S for
// (B,H,S,D) = (8,16,4096,128) fp32. Memory-bound (512 MB @ 23.3 TB/s ~ 22us).
// Implemented as chained 16x16 scan tiles on the matrix pipe:
//   Y_tile = T * X_tile + outer(g^{i+1}, carry),  T[i][j] = g^(i-j) (lower tri)
// using 4x V_WMMA_F32_16X16X4_F32 per tile, carry extracted from row 15.

typedef __attribute__((ext_vector_type(2))) float v2f;
typedef __attribute__((ext_vector_type(8))) float v8f;

#define S_LEN 4096
#define D_DIM 128
#define H_DIM 16
#define NT    (S_LEN / 16)   // 256 tiles per scan chain
#define DEPTH 8              // software-pipeline depth (8 KB in flight / wave)

__device__ __forceinline__ float powg(float g, int e) {
  // g^e for e in [0,16], branchless (done once at wave start)
  float r = 1.0f;
#pragma unroll
  for (int i = 0; i < 16; ++i) r = (i < e) ? r * g : r;
  return r;
}

__global__ __launch_bounds__(256)
void disc_cumsum_wmma(const float* __restrict__ x,
                      const float* __restrict__ gamma,
                      float* __restrict__ y) {
  const int tid  = blockIdx.x * blockDim.x + threadIdx.x;
  const int w    = tid >> 5;            // global wave id, 0..1023
  const int lane = threadIdx.x & 31;
  const int bh   = w >> 3;              // (b,h) pair, 0..127
  const int dg   = w & 7;               // 16-wide d-group, 0..7
  const int h    = bh & (H_DIM - 1);
  const int n    = lane & 15;           // column within d-group / row for A
  const int hi   = lane >> 4;           // half-wave select

  const float g = gamma[h];

  // A matrix: 4 blocks of 16x4 from T[i][j] = g^(i-j), lower triangular.
  // 32-bit A layout: lanes 0-15 row M=lane (VGPR0:K=0, VGPR1:K=1),
  //                  lanes 16-31 row M=lane-16 (VGPR0:K=2, VGPR1:K=3).
  v2f a[4];
#pragma unroll
  for (int k = 0; k < 4; ++k) {
    const int j0 = 4 * k + 2 * hi;
    a[k].x = (j0     <= n) ? powg(g, n - j0)     : 0.0f;
    a[k].y = (j0 + 1 <= n) ? powg(g, n - j0 - 1) : 0.0f;
  }

  // Carry-injection factors for the C accumulator: f[v] = g^(row+1),
  // row = v + 8*hi per the 16x16 f32 C/D layout.
  v8f f;
#pragma unroll
  for (int v = 0; v < 8; ++v) f[v] = powg(g, v + 8 * hi + 1);

  // Base pointers. Load rows for B-block k are {4k+2*hi, 4k+2*hi+1};
  // store rows for C VGPR v are {v + 8*hi}. Fold the hi-dependent row in.
  const size_t mat = (size_t)bh * (size_t)(S_LEN * D_DIM);
  const int    col = dg * 16 + n;
  const float* xb  = x + mat + col + (size_t)(2 * hi) * D_DIM;
  float*       yb  = y + mat + col + (size_t)(8 * hi) * D_DIM;

  v8f   c;
  float carry = 0.0f;
  v2f   buf[DEPTH][4];

  auto load_tile = [&](int t, v2f (&bb)[4]) {
    const float* p = xb + (size_t)t * (16 * D_DIM);
#pragma unroll
    for (int k = 0; k < 4; ++k) {             // B 4x16 block k: rows 4k..4k+3
      bb[k].x = p[(4 * k) * D_DIM];           // K = 4k + 2*hi
      bb[k].y = p[(4 * k + 1) * D_DIM];       // K = 4k + 2*hi + 1
    }
  };

  auto consume = [&](int t, v2f (&bb)[4]) {
    // C = outer(g^{i+1}, carry)
#pragma unroll
    for (int v = 0; v < 8; ++v) c[v] = f[v] * carry;
    // Y = T*X + C as 4 chained 16x16x4 f32 WMMAs
    c = __builtin_amdgcn_wmma_f32_16x16x4_f32(false, a[0], false, bb[0],
                                              (short)0, c, false, false);
    c = __builtin_amdgcn_wmma_f32_16x16x4_f32(false, a[1], false, bb[1],
                                              (short)0, c, false, false);
    c = __builtin_amdgcn_wmma_f32_16x16x4_f32(false, a[2], false, bb[2],
                                              (short)0, c, false, false);
    c = __builtin_amdgcn_wmma_f32_16x16x4_f32(false, a[3], false, bb[3],
                                              (short)0, c, false, false);
    // new carry = row 15 of Y (VGPR 7, lanes 16-31), broadcast per column
    carry = __shfl(c[7], 16 + n, 32);
    float* q = yb + (size_t)t * (16 * D_DIM);
#pragma unroll
    for (int v = 0; v < 8; ++v) q[v * D_DIM] = c[v];  // row v + 8*hi
  };

  // ---- software pipeline: DEPTH tiles of loads in flight ----
#pragma unroll
  for (int p = 0; p < DEPTH; ++p) load_tile(p, buf[p]);

  for (int t = 0; t < NT - DEPTH; t += DEPTH) {
#pragma unroll
    for (int u = 0; u < DEPTH; ++u) {
      consume(t + u, buf[u]);
      load_tile(t + u + DEPTH, buf[u]);
    }
  }
#pragma unroll
  for (int u = 0; u < DEPTH; ++u) consume(NT - DEPTH + u, buf[u]);
}

extern "C" void kernel_launch(void* const* d_in, const int* in_sizes, int n_in,
                              void* d_out, int out_size, void* d_ws, size_t ws_size,
                              hipStream_t stream) {
  const float* x     = (const float*)d_in[0];   // tensor (B,H,S,D) fp32
  const float* gamma = (const float*)d_in[1];   // (H,) fp32
  float*       out   = (float*)d_out;
  (void)in_sizes; (void)n_in; (void)out_size; (void)d_ws; (void)ws_size;

  // 1024 waves = 8 waves/block x 128 blocks; each block owns one (b,h) and
  // its 8 d-groups, so concurrent waves read fully contiguous 512B rows.
  disc_cumsum_wmma<<<dim3(128), dim3(256), 0, stream>>>(x, gamma, out);
}